// Sinkhorn_71683004170748
// MI455X (gfx1250) — compile-verified
//
#include <hip/hip_runtime.h>
#include <cstdint>

// Sinkhorn via dual potentials, matrix LDS-resident, READ-ONLY, and pre-scaled
// to base-2 units (x2 = x*log2(e)) so the hot loop uses raw v_exp_f32 / v_log_f32:
//
//   colOff2[c] = L2SE_r( x2[r][c] - rowOff2[r] )      (L2SE = log2-sum-exp2)
//   rowOff2[r] = L2SE_c( x2[r][c] - colOff2[c] )
//   out        = exp2( x2 - rowOff2[r] - colOff2[c] )
//
// One workgroup (256 threads = 8 wave32) per 256x256 matrix (512 matrices).
// LDS pitch 260 floats: row base = t*1040 B (16B aligned -> b128 at full bank BW),
// column word addr r*260+c -> bank (4r+c)%64 -> conflict-free scalar column walk.

#define SINK_N     256
#define SINK_LDW   260
#define SINK_ITERS 20

typedef float vfloat4 __attribute__((ext_vector_type(4)));

__global__ __launch_bounds__(SINK_N, 1)
void Sinkhorn_71683004170748_kernel(const float* __restrict__ x,
                                    float* __restrict__ out,
                                    int n_mat) {
    __shared__ __align__(16) float smem[SINK_N * SINK_LDW];
    __shared__ __align__(16) float rowOff2[SINK_N];
    __shared__ __align__(16) float colOff2[SINK_N];

    const int t = (int)threadIdx.x;           // 0..255
    const int b = (int)blockIdx.x;            // matrix id
    if (b >= n_mat) return;

    const size_t mat_off = (size_t)b * (SINK_N * SINK_N);
    const float  LOG2E   = 1.4426950408889634f;

    // ---------------- async global -> LDS fill (CDNA5 ASYNCcnt path) -------------
    // Thread t DMAs its row t: 64 x b128, both sides 16B aligned.
    {
        const float* g = x + mat_off + (size_t)t * SINK_N;
        #pragma unroll
        for (int k = 0; k < SINK_N / 4; ++k) {
            unsigned laddr = (unsigned)(uintptr_t)(&smem[t * SINK_LDW + 4 * k]);
            unsigned long long gaddr = (unsigned long long)(uintptr_t)(g + 4 * k);
            asm volatile("global_load_async_to_lds_b128 %0, %1, off"
                         :: "v"(laddr), "v"(gaddr)
                         : "memory");
        }
    }
    rowOff2[t] = 0.0f;                         // initial potentials (base-2 units)
    asm volatile("s_wait_asynccnt 0" ::: "memory");

    // One-time scale to base-2 units. Thread t only touches the row its own DMA
    // wrote (already waited on), so no barrier is needed before this.
    {
        vfloat4* rowp = reinterpret_cast<vfloat4*>(&smem[t * SINK_LDW]);
        #pragma unroll 8
        for (int k = 0; k < SINK_N / 4; ++k)
            rowp[k] = rowp[k] * LOG2E;
    }
    __syncthreads();

    #pragma unroll 1
    for (int it = 0; it < SINK_ITERS; ++it) {
        // ---- column pass: thread t owns column t --------------------------------
        // rowOff2[r..r+3] fetched as one wave-uniform broadcast b128.
        {
            const float* colp = &smem[t];

            float m0 = -INFINITY, m1 = -INFINITY, m2 = -INFINITY, m3 = -INFINITY;
            #pragma unroll 4
            for (int r = 0; r < SINK_N; r += 4) {
                const vfloat4 ro = *reinterpret_cast<const vfloat4*>(&rowOff2[r]);
                m0 = fmaxf(m0, colp[(r + 0) * SINK_LDW] - ro.x);
                m1 = fmaxf(m1, colp[(r + 1) * SINK_LDW] - ro.y);
                m2 = fmaxf(m2, colp[(r + 2) * SINK_LDW] - ro.z);
                m3 = fmaxf(m3, colp[(r + 3) * SINK_LDW] - ro.w);
            }
            const float m = fmaxf(fmaxf(m0, m1), fmaxf(m2, m3));

            float s0 = 0.f, s1 = 0.f, s2 = 0.f, s3 = 0.f;
            #pragma unroll 4
            for (int r = 0; r < SINK_N; r += 4) {
                const vfloat4 ro = *reinterpret_cast<const vfloat4*>(&rowOff2[r]);
                s0 += __builtin_amdgcn_exp2f((colp[(r + 0) * SINK_LDW] - ro.x) - m);
                s1 += __builtin_amdgcn_exp2f((colp[(r + 1) * SINK_LDW] - ro.y) - m);
                s2 += __builtin_amdgcn_exp2f((colp[(r + 2) * SINK_LDW] - ro.z) - m);
                s3 += __builtin_amdgcn_exp2f((colp[(r + 3) * SINK_LDW] - ro.w) - m);
            }
            colOff2[t] = m + __builtin_amdgcn_logf((s0 + s1) + (s2 + s3));
        }
        __syncthreads();

        // ---- row pass: thread t owns row t (b128 matrix reads) ------------------
        {
            const vfloat4* rowp = reinterpret_cast<const vfloat4*>(&smem[t * SINK_LDW]);
            const vfloat4* cofs = reinterpret_cast<const vfloat4*>(colOff2);

            float mx = -INFINITY, my = -INFINITY, mz = -INFINITY, mw = -INFINITY;
            #pragma unroll 4
            for (int k = 0; k < SINK_N / 4; ++k) {
                const vfloat4 v = rowp[k] - cofs[k];
                mx = fmaxf(mx, v.x); my = fmaxf(my, v.y);
                mz = fmaxf(mz, v.z); mw = fmaxf(mw, v.w);
            }
            const float m = fmaxf(fmaxf(mx, my), fmaxf(mz, mw));

            float sx = 0.f, sy = 0.f, sz = 0.f, sw = 0.f;
            #pragma unroll 4
            for (int k = 0; k < SINK_N / 4; ++k) {
                const vfloat4 v = (rowp[k] - cofs[k]) - m;
                sx += __builtin_amdgcn_exp2f(v.x);
                sy += __builtin_amdgcn_exp2f(v.y);
                sz += __builtin_amdgcn_exp2f(v.z);
                sw += __builtin_amdgcn_exp2f(v.w);
            }
            rowOff2[t] = m + __builtin_amdgcn_logf((sx + sy) + (sz + sw));
        }
        __syncthreads();
    }

    // ---------------- final: out = exp2(x2 - rowOff2[r] - colOff2[c]) ------------
    // Thread t owns row t. rowOff2[t] is this thread's own last write (no barrier
    // needed); colOff2 is fenced by the final __syncthreads above.
    {
        const vfloat4* rowp = reinterpret_cast<const vfloat4*>(&smem[t * SINK_LDW]);
        const vfloat4* cofs = reinterpret_cast<const vfloat4*>(colOff2);
        const float    ro   = rowOff2[t];
        float* dst = out + mat_off + (size_t)t * SINK_N;

        #pragma unroll 4
        for (int k = 0; k < SINK_N / 4; ++k) {
            const vfloat4 v = (rowp[k] - cofs[k]) - ro;
            vfloat4 e;
            e.x = __builtin_amdgcn_exp2f(v.x);
            e.y = __builtin_amdgcn_exp2f(v.y);
            e.z = __builtin_amdgcn_exp2f(v.z);
            e.w = __builtin_amdgcn_exp2f(v.w);
            __builtin_nontemporal_store(e, reinterpret_cast<vfloat4*>(dst + 4 * k));
        }
    }
}

extern "C" void kernel_launch(void* const* d_in, const int* in_sizes, int n_in,
                              void* d_out, int out_size, void* d_ws, size_t ws_size,
                              hipStream_t stream) {
    (void)d_ws; (void)ws_size; (void)out_size; (void)n_in;
    const float* x  = (const float*)d_in[0];
    float*       o  = (float*)d_out;
    const int n_mat = in_sizes[0] / (SINK_N * SINK_N);   // 64*8 = 512

    Sinkhorn_71683004170748_kernel<<<dim3((unsigned)n_mat), dim3(SINK_N), 0, stream>>>(x, o, n_mat);
}